// MessagePassingLayer_89721866813784
// MI455X (gfx1250) — compile-verified
//
#include <hip/hip_runtime.h>
#include <math.h>

// ---------------- problem constants ----------------
#define BB       4096
#define NN       22
#define EE       42
#define DD       128
#define EDIM     32
#define FF       2048
#define BN       (BB * NN)      // 90112
#define BE       (BB * EE)      // 172032
#define CIN      (2 * DD + EDIM) // 288

typedef __attribute__((ext_vector_type(2))) float v2f;
typedef __attribute__((ext_vector_type(8))) float v8f;

// D = A(16x4 f32) * B(4x16 f32) + C, wave32, 2048 FLOP per instruction
__device__ __forceinline__ v8f wmma4(float a0, float a1, float b0, float b1, v8f c) {
    v2f a; a.x = a0; a.y = a1;
    v2f b; b.x = b0; b.y = b1;
    return __builtin_amdgcn_wmma_f32_16x16x4_f32(false, a, false, b, (short)0, c, false, false);
}

__device__ __forceinline__ float sigmoidf_(float x) { return 1.0f / (1.0f + __expf(-x)); }
__device__ __forceinline__ float gelu_exact(float x) {
    return 0.5f * x * (1.0f + erff(x * 0.70710678118654752f));
}

// ---------------- Kernel 1/5: LayerNorm, one wave per 128-wide row ----------------
__global__ void ln_kernel(const float* __restrict__ in, const float* __restrict__ gamma,
                          const float* __restrict__ beta, float* __restrict__ out, int rows) {
    int wave = threadIdx.x >> 5;
    int lane = threadIdx.x & 31;
    int row  = blockIdx.x * 8 + wave;
    if (row >= rows) return;
    float4 v = ((const float4*)(in + (size_t)row * DD))[lane];
    float s  = v.x + v.y + v.z + v.w;
    float sq = v.x * v.x + v.y * v.y + v.z * v.z + v.w * v.w;
    #pragma unroll
    for (int m = 16; m >= 1; m >>= 1) {
        s  += __shfl_xor(s, m, 32);
        sq += __shfl_xor(sq, m, 32);
    }
    float mean = s * (1.0f / DD);
    float var  = sq * (1.0f / DD) - mean * mean;
    float inv  = rsqrtf(var + 1e-5f);
    float4 g4 = ((const float4*)gamma)[lane];
    float4 b4 = ((const float4*)beta)[lane];
    float4 o;
    o.x = (v.x - mean) * inv * g4.x + b4.x;
    o.y = (v.y - mean) * inv * g4.y + b4.y;
    o.z = (v.z - mean) * inv * g4.z + b4.z;
    o.w = (v.w - mean) * inv * g4.w + b4.w;
    ((float4*)(out + (size_t)row * DD))[lane] = o;
}

// ---------------- Kernel 2: fused edge MLP (288 -> 128 relu -> 32), 32 rows/block ----------------
// Two 16-row A tiles per block: every weight fragment fetched once feeds two WMMAs.
#define CSTR 292
#define HSTR 132
__global__ void edge_mlp_kernel(const float* __restrict__ normed, const float* __restrict__ edge_attr,
                                const int* __restrict__ si, const int* __restrict__ ti,
                                const float* __restrict__ ew1, const float* __restrict__ eb1,
                                const float* __restrict__ ew2, const float* __restrict__ eb2,
                                float* __restrict__ msg_out) {
    __shared__ float comb[32 * CSTR];
    __shared__ float hs[32 * HSTR];
    const int tile = blockIdx.x;
    const int tid  = threadIdx.x;

    // stage combined = [normed[src], normed[tgt], edge_attr] : 32 rows x 288
    {
        int rowi = tid >> 3;                // 0..31
        int lr   = tid & 7;
        int r    = tile * 32 + rowi;        // global edge row = b*EE + e
        int b    = r / EE;
        int e    = r % EE;
        const float* srcp = normed + ((size_t)b * NN + si[e]) * DD;
        const float* tgtp = normed + ((size_t)b * NN + ti[e]) * DD;
        const float* eap  = edge_attr + (size_t)r * EDIM;
        for (int k = lr; k < CIN; k += 8) {
            float v;
            if (k < DD)          v = srcp[k];
            else if (k < 2 * DD) v = tgtp[k - DD];
            else                 v = eap[k - 2 * DD];
            comb[rowi * CSTR + k] = v;
        }
    }
    __syncthreads();

    const int wave = tid >> 5, lane = tid & 31;
    const int hl = lane >> 4, mn = lane & 15;
    // layer 1: each wave owns output cols [wave*16, wave*16+16) for both row tiles
    {
        int n0 = wave * 16;
        v8f acc0, acc1;
        float bias = eb1[n0 + mn];
        #pragma unroll
        for (int i = 0; i < 8; i++) { acc0[i] = bias; acc1[i] = bias; }
        for (int k0 = 0; k0 < CIN; k0 += 4) {
            int ka = k0 + 2 * hl;
            float b0 = ew1[(size_t)ka * (4 * EDIM) + n0 + mn];
            float b1 = ew1[(size_t)(ka + 1) * (4 * EDIM) + n0 + mn];
            float a0 = comb[mn * CSTR + ka];
            float a1 = comb[mn * CSTR + ka + 1];
            float a2 = comb[(16 + mn) * CSTR + ka];
            float a3 = comb[(16 + mn) * CSTR + ka + 1];
            acc0 = wmma4(a0, a1, b0, b1, acc0);
            acc1 = wmma4(a2, a3, b0, b1, acc1);
        }
        #pragma unroll
        for (int i = 0; i < 8; i++) {
            int m = i + 8 * hl;
            hs[m * HSTR + n0 + mn]        = fmaxf(acc0[i], 0.0f);
            hs[(16 + m) * HSTR + n0 + mn] = fmaxf(acc1[i], 0.0f);
        }
    }
    __syncthreads();
    // layer 2: 32 output cols -> 2 column tiles, waves 0 and 1, both row tiles each
    if (wave < 2) {
        int n0 = wave * 16;
        v8f acc0, acc1;
        float bias = eb2[n0 + mn];
        #pragma unroll
        for (int i = 0; i < 8; i++) { acc0[i] = bias; acc1[i] = bias; }
        for (int k0 = 0; k0 < DD; k0 += 4) {
            int ka = k0 + 2 * hl;
            float b0 = ew2[(size_t)ka * EDIM + n0 + mn];
            float b1 = ew2[(size_t)(ka + 1) * EDIM + n0 + mn];
            float a0 = hs[mn * HSTR + ka];
            float a1 = hs[mn * HSTR + ka + 1];
            float a2 = hs[(16 + mn) * HSTR + ka];
            float a3 = hs[(16 + mn) * HSTR + ka + 1];
            acc0 = wmma4(a0, a1, b0, b1, acc0);
            acc1 = wmma4(a2, a3, b0, b1, acc1);
        }
        #pragma unroll
        for (int i = 0; i < 8; i++) {
            int m = i + 8 * hl;
            size_t r = (size_t)tile * 32 + m;
            msg_out[r * EDIM + n0 + mn]        = acc0[i];
            msg_out[(r + 16) * EDIM + n0 + mn] = acc1[i];
        }
    }
}

// ---------------- Kernel 3: aggregation (22x42 @ 42x32) + projection (32 -> 128) ----------------
__global__ void agg_proj_kernel(const float* __restrict__ aggm, const float* __restrict__ msg,
                                const float* __restrict__ pw, const float* __restrict__ pb,
                                float* __restrict__ projected) {
    __shared__ float msgs[EE * EDIM];   // 42*32
    __shared__ float am[NN * EE];       // 22*42
    __shared__ float agl[NN * EDIM];    // 22*32
    const int b = blockIdx.x, tid = threadIdx.x;
    const float* mb = msg + (size_t)b * EE * EDIM;
    for (int i = tid; i < EE * EDIM; i += 256) msgs[i] = mb[i];
    for (int i = tid; i < NN * EE; i += 256) am[i] = aggm[i];
    __syncthreads();
    for (int idx = tid; idx < NN * EDIM; idx += 256) {
        int n = idx / EDIM, c = idx % EDIM;
        float s = 0.0f;
        #pragma unroll 6
        for (int e = 0; e < EE; e++) s += am[n * EE + e] * msgs[e * EDIM + c];
        agl[idx] = s;
    }
    __syncthreads();
    for (int idx = tid; idx < NN * DD; idx += 256) {
        int n = idx / DD, d = idx % DD;
        float s = pb[d];
        #pragma unroll 8
        for (int c = 0; c < EDIM; c++) s += agl[n * EDIM + c] * pw[c * DD + d];
        projected[((size_t)b * NN + n) * DD + d] = s;
    }
}

// ---------------- Kernel 4: fused GRU update (16 rows/block) ----------------
#define NMSTR 264
__global__ void gru_kernel(const float* __restrict__ node, const float* __restrict__ normed,
                           const float* __restrict__ projected,
                           const float* __restrict__ rw, const float* __restrict__ rb,
                           const float* __restrict__ uw, const float* __restrict__ ub,
                           const float* __restrict__ cw, const float* __restrict__ cb,
                           float* __restrict__ x_out) {
    __shared__ float nm[16 * NMSTR];   // [normed, projected] 16 x 256
    __shared__ float rc[16 * NMSTR];   // [reset*normed, projected]
    __shared__ float ug[16 * DD];      // update gate
    const int tile = blockIdx.x, tid = threadIdx.x;
    {
        int rowi = tid >> 4, lr = tid & 15;
        size_t r = (size_t)tile * 16 + rowi;
        for (int k = lr; k < 2 * DD; k += 16) {
            float v = (k < DD) ? normed[r * DD + k] : projected[r * DD + k - DD];
            nm[rowi * NMSTR + k] = v;
            if (k >= DD) rc[rowi * NMSTR + k] = v;   // rc second half = projected
        }
    }
    __syncthreads();
    const int wave = tid >> 5, lane = tid & 31;
    const int hl = lane >> 4, mn = lane & 15;
    const int n0 = wave * 16;
    // reset & update gates: K = 256 (A fragment reused for both gates)
    {
        v8f ra, ua;
        float rbv = rb[n0 + mn], ubv = ub[n0 + mn];
        #pragma unroll
        for (int i = 0; i < 8; i++) { ra[i] = rbv; ua[i] = ubv; }
        for (int k0 = 0; k0 < 2 * DD; k0 += 4) {
            int ka = k0 + 2 * hl;
            float a0 = nm[mn * NMSTR + ka];
            float a1 = nm[mn * NMSTR + ka + 1];
            float r0 = rw[(size_t)ka * DD + n0 + mn];
            float r1 = rw[(size_t)(ka + 1) * DD + n0 + mn];
            float u0 = uw[(size_t)ka * DD + n0 + mn];
            float u1 = uw[(size_t)(ka + 1) * DD + n0 + mn];
            ra = wmma4(a0, a1, r0, r1, ra);
            ua = wmma4(a0, a1, u0, u1, ua);
        }
        #pragma unroll
        for (int i = 0; i < 8; i++) {
            int m = i + 8 * hl;
            float rv = sigmoidf_(ra[i]);
            float uv = sigmoidf_(ua[i]);
            rc[m * NMSTR + n0 + mn] = rv * nm[m * NMSTR + n0 + mn];
            ug[m * DD + n0 + mn]    = uv;
        }
    }
    __syncthreads();
    // candidate: K = 256 over rc
    {
        v8f ca;
        float cbv = cb[n0 + mn];
        #pragma unroll
        for (int i = 0; i < 8; i++) ca[i] = cbv;
        for (int k0 = 0; k0 < 2 * DD; k0 += 4) {
            int ka = k0 + 2 * hl;
            float a0 = rc[mn * NMSTR + ka];
            float a1 = rc[mn * NMSTR + ka + 1];
            float c0 = cw[(size_t)ka * DD + n0 + mn];
            float c1 = cw[(size_t)(ka + 1) * DD + n0 + mn];
            ca = wmma4(a0, a1, c0, c1, ca);
        }
        #pragma unroll
        for (int i = 0; i < 8; i++) {
            int m = i + 8 * hl;
            int col = n0 + mn;
            size_t r = (size_t)tile * 16 + m;
            float cand = tanhf(ca[i]);
            float uv = ug[m * DD + col];
            float nv = nm[m * NMSTR + col];
            x_out[r * DD + col] = node[r * DD + col] + (1.0f - uv) * nv + uv * cand;
        }
    }
}

// ---------------- Kernel 6: fused FFN (128 -> 2048 gelu -> 128) + residual, 32 rows/block ----------------
#define TSTR 132
__global__ void ffn_kernel(const float* __restrict__ n2, const float* __restrict__ x,
                           const float* __restrict__ fw1, const float* __restrict__ fb1,
                           const float* __restrict__ fw2, const float* __restrict__ fb2,
                           float* __restrict__ out) {
    __shared__ float nt[32 * TSTR];
    __shared__ float g[32 * TSTR];
    const int tile = blockIdx.x, tid = threadIdx.x;
    const size_t r0 = (size_t)tile * 32;
    {
        int rowi = tid >> 3, lr = tid & 7;   // 32 rows, 8 threads each
        for (int k = lr; k < DD; k += 8) nt[rowi * TSTR + k] = n2[(r0 + rowi) * DD + k];
    }
    __syncthreads();
    const int wave = tid >> 5, lane = tid & 31;
    const int hl = lane >> 4, mn = lane & 15;
    const int n0 = wave * 16;
    v8f acc0, acc1;
    {
        float b2 = fb2[n0 + mn];
        #pragma unroll
        for (int i = 0; i < 8; i++) { acc0[i] = b2; acc1[i] = b2; }
    }
    for (int fc = 0; fc < FF; fc += 128) {
        // first matmul chunk: 32 x 128 of hidden; weight fragment feeds 2 WMMAs
        int f0 = fc + n0;
        v8f h0, h1;
        float b1 = fb1[f0 + mn];
        #pragma unroll
        for (int i = 0; i < 8; i++) { h0[i] = b1; h1[i] = b1; }
        for (int k0 = 0; k0 < DD; k0 += 4) {
            int ka = k0 + 2 * hl;
            float w0 = fw1[(size_t)ka * FF + f0 + mn];
            float w1 = fw1[(size_t)(ka + 1) * FF + f0 + mn];
            float a0 = nt[mn * TSTR + ka];
            float a1 = nt[mn * TSTR + ka + 1];
            float a2 = nt[(16 + mn) * TSTR + ka];
            float a3 = nt[(16 + mn) * TSTR + ka + 1];
            h0 = wmma4(a0, a1, w0, w1, h0);
            h1 = wmma4(a2, a3, w0, w1, h1);
        }
        __syncthreads();   // previous chunk's readers of g are done
        #pragma unroll
        for (int i = 0; i < 8; i++) {
            int m = i + 8 * hl;
            g[m * TSTR + n0 + mn]        = gelu_exact(h0[i]);
            g[(16 + m) * TSTR + n0 + mn] = gelu_exact(h1[i]);
        }
        __syncthreads();
        // second matmul chunk: accumulate both out tiles over this K-chunk of 128
        for (int k0 = 0; k0 < DD; k0 += 4) {
            int ka = k0 + 2 * hl;
            float w0 = fw2[(size_t)(fc + ka) * DD + n0 + mn];
            float w1 = fw2[(size_t)(fc + ka + 1) * DD + n0 + mn];
            float a0 = g[mn * TSTR + ka];
            float a1 = g[mn * TSTR + ka + 1];
            float a2 = g[(16 + mn) * TSTR + ka];
            float a3 = g[(16 + mn) * TSTR + ka + 1];
            acc0 = wmma4(a0, a1, w0, w1, acc0);
            acc1 = wmma4(a2, a3, w0, w1, acc1);
        }
    }
    #pragma unroll
    for (int i = 0; i < 8; i++) {
        int m = i + 8 * hl;
        int col = n0 + mn;
        size_t ra = r0 + m;
        size_t rb = r0 + 16 + m;
        out[ra * DD + col] = x[ra * DD + col] + acc0[i];
        out[rb * DD + col] = x[rb * DD + col] + acc1[i];
    }
}

// ---------------- launch ----------------
extern "C" void kernel_launch(void* const* d_in, const int* in_sizes, int n_in,
                              void* d_out, int out_size, void* d_ws, size_t ws_size,
                              hipStream_t stream) {
    const float* node = (const float*)d_in[0];
    const float* edge_attr = (const float*)d_in[1];
    const float* aggm = (const float*)d_in[2];
    const int*   si   = (const int*)d_in[3];
    const int*   ti   = (const int*)d_in[4];
    const float* ew1 = (const float*)d_in[5];  const float* eb1 = (const float*)d_in[6];
    const float* ew2 = (const float*)d_in[7];  const float* eb2 = (const float*)d_in[8];
    const float* pw  = (const float*)d_in[9];  const float* pb  = (const float*)d_in[10];
    const float* rw  = (const float*)d_in[11]; const float* rb  = (const float*)d_in[12];
    const float* uw  = (const float*)d_in[13]; const float* ub  = (const float*)d_in[14];
    const float* cw  = (const float*)d_in[15]; const float* cb  = (const float*)d_in[16];
    const float* g1  = (const float*)d_in[17]; const float* be1 = (const float*)d_in[18];
    const float* g2  = (const float*)d_in[19]; const float* be2 = (const float*)d_in[20];
    const float* fw1 = (const float*)d_in[21]; const float* fb1 = (const float*)d_in[22];
    const float* fw2 = (const float*)d_in[23]; const float* fb2 = (const float*)d_in[24];

    float* out     = (float*)d_out;
    float* msg_out = out + (size_t)BN * DD;            // second output, [B,E,32]

    float* ws        = (float*)d_ws;
    float* normed    = ws;                              // reused as n2 after GRU
    float* projected = ws + (size_t)BN * DD;
    float* xbuf      = ws + 2 * (size_t)BN * DD;

    ln_kernel<<<BN / 8, 256, 0, stream>>>(node, g1, be1, normed, BN);
    edge_mlp_kernel<<<BE / 32, 256, 0, stream>>>(normed, edge_attr, si, ti,
                                                 ew1, eb1, ew2, eb2, msg_out);
    agg_proj_kernel<<<BB, 256, 0, stream>>>(aggm, msg_out, pw, pb, projected);
    gru_kernel<<<BN / 16, 256, 0, stream>>>(node, normed, projected,
                                            rw, rb, uw, ub, cw, cb, xbuf);
    ln_kernel<<<BN / 8, 256, 0, stream>>>(xbuf, g2, be2, normed /* n2 */, BN);
    ffn_kernel<<<BN / 32, 256, 0, stream>>>(normed, xbuf, fw1, fb1, fw2, fb2, out);
}